// SubsetOperator_3118146257589
// MI455X (gfx1250) — compile-verified
//
#include <hip/hip_runtime.h>

// Problem constants (match reference)
#define NCOLS   16384
#define NROWS   2048
#define KITERS  16
#define BLOCK   512                    // 16 wave32s
#define NWAVES  (BLOCK / 32)           // 16
#define VPT     (NCOLS / BLOCK)        // 32 elements per thread, in registers
#define NV4     (VPT / 4)              // 8 float4 chunks per thread
#define RPB     4                      // rows per block (enables DMA prefetch overlap)
#define NBLOCKS (NROWS / RPB)          // 512 blocks

#define L2E      1.44269504088896340736f   // log2(e)
#define EPS_TINY 1.17549435e-38f           // np.finfo(float32).tiny

// Fast hardware transcendentals: v_exp_f32 / v_log_f32 (base-2)
#if defined(__HIP_DEVICE_COMPILE__) && __has_builtin(__builtin_amdgcn_exp2f)
#define EXP2F(x) __builtin_amdgcn_exp2f(x)
#else
#define EXP2F(x) exp2f(x)
#endif
#if defined(__HIP_DEVICE_COMPILE__) && __has_builtin(__builtin_amdgcn_logf)
#define LOG2F(x) __builtin_amdgcn_logf(x)
#else
#define LOG2F(x) log2f(x)
#endif

// CDNA5 async global->LDS DMA path (gfx1250). Param type per compiler
// diagnostic: (global v4i*, lds v4i*, imm offset, imm cpol).
#if defined(__gfx1250__) && __has_builtin(__builtin_amdgcn_global_load_async_to_lds_b128) && __has_builtin(__builtin_amdgcn_s_wait_asynccnt)
#define USE_ASYNC 1
#else
#define USE_ASYNC 0
#endif

#if USE_ASYNC
typedef int v4i __attribute__((ext_vector_type(4)));
#define AS1 __attribute__((address_space(1)))
#define AS3 __attribute__((address_space(3)))
__device__ __forceinline__ void async_cp_b128(const float* gsrc, float* ldst) {
  __builtin_amdgcn_global_load_async_to_lds_b128((AS1 v4i*)gsrc, (AS3 v4i*)ldst,
                                                 /*imm offset*/0, /*cpol*/0);
}
#endif

__global__ __launch_bounds__(BLOCK) void subset_topk_kernel(
    const float* __restrict__ scores,
    const float* __restrict__ gnoise,
    float* __restrict__ out)
{
  const int t    = threadIdx.x;
  const int lane = t & 31;
  const int wid  = t >> 5;

  __shared__ float red[2 * NWAVES];        // [0,16): wave maxes  [16,32): wave sums
  extern __shared__ float stage[];         // 2*NCOLS floats: scores row | g row
  float* stage_s = stage;
  float* stage_g = stage + NCOLS;

  const int row0 = blockIdx.x * RPB;

#if USE_ASYNC
  { // prefetch first row into LDS stage (each wave stages only its own region)
    const float* gs = scores + (size_t)row0 * NCOLS;
    const float* gg = gnoise + (size_t)row0 * NCOLS;
#pragma unroll
    for (int v = 0; v < NV4; ++v) {
      const int e = (v * BLOCK + t) * 4;
      async_cp_b128(gs + e, stage_s + e);
      async_cp_b128(gg + e, stage_g + e);
    }
  }
#endif

  for (int r = 0; r < RPB; ++r) {
    const size_t rbase = (size_t)(row0 + r) * NCOLS;
    float u[VPT], kh[VPT], ex[VPT];

#if USE_ASYNC
    __builtin_amdgcn_s_wait_asynccnt(0);   // DMA of this row's stage complete
#pragma unroll
    for (int v = 0; v < NV4; ++v) {
      const int e = (v * BLOCK + t) * 4;
      const float4 a = *(const float4*)(stage_s + e);
      const float4 b = *(const float4*)(stage_g + e);
      u[4*v+0] = (a.x + b.x) * L2E;
      u[4*v+1] = (a.y + b.y) * L2E;
      u[4*v+2] = (a.z + b.z) * L2E;
      u[4*v+3] = (a.w + b.w) * L2E;
    }
    if (r + 1 < RPB) {
      // ensure our stage reads retired before the DMA engine overwrites them
      asm volatile("s_wait_dscnt 0" ::: "memory");
      const float* gs = scores + rbase + NCOLS;
      const float* gg = gnoise + rbase + NCOLS;
#pragma unroll
      for (int v = 0; v < NV4; ++v) {
        const int e = (v * BLOCK + t) * 4;
        async_cp_b128(gs + e, stage_s + e);
        async_cp_b128(gg + e, stage_g + e);
      }
    }
#else
#pragma unroll
    for (int v = 0; v < NV4; ++v) {
      const int e = (v * BLOCK + t) * 4;
      const float4 a = *(const float4*)(scores + rbase + e);
      const float4 b = *(const float4*)(gnoise + rbase + e);
      u[4*v+0] = (a.x + b.x) * L2E;
      u[4*v+1] = (a.y + b.y) * L2E;
      u[4*v+2] = (a.z + b.z) * L2E;
      u[4*v+3] = (a.w + b.w) * L2E;
    }
#endif

#pragma unroll
    for (int k = 0; k < VPT; ++k) kh[k] = 0.0f;

    for (int it = 0; it < KITERS; ++it) {
      // ---- phase A: row max of u (log2 domain) ----
      float m = -3.402823466e38f;
#pragma unroll
      for (int k = 0; k < VPT; ++k) m = fmaxf(m, u[k]);
#pragma unroll
      for (int off = 16; off > 0; off >>= 1)
        m = fmaxf(m, __shfl_xor(m, off, 32));
      if (lane == 0) red[wid] = m;
      __syncthreads();
      m = red[lane & (NWAVES - 1)];
#pragma unroll
      for (int off = NWAVES / 2; off > 0; off >>= 1)
        m = fmaxf(m, __shfl_xor(m, off, 32));

      // ---- phase B: Z = sum exp2(u - m) ----
      float z = 0.0f;
#pragma unroll
      for (int k = 0; k < VPT; ++k) { ex[k] = EXP2F(u[k] - m); z += ex[k]; }
#pragma unroll
      for (int off = 16; off > 0; off >>= 1)
        z += __shfl_xor(z, off, 32);
      if (lane == 0) red[NWAVES + wid] = z;
      __syncthreads();
      z = red[NWAVES + (lane & (NWAVES - 1))];
#pragma unroll
      for (int off = NWAVES / 2; off > 0; off >>= 1)
        z += __shfl_xor(z, off, 32);

      // ---- phase C: khot += p ; u += log2(max(1-p, tiny)) ----
      const float rz = __builtin_amdgcn_rcpf(z);
      if (it + 1 < KITERS) {
#pragma unroll
        for (int k = 0; k < VPT; ++k) {
          const float p = ex[k] * rz;
          kh[k] += p;
          u[k]  += LOG2F(fmaxf(1.0f - p, EPS_TINY));
        }
      } else {
#pragma unroll
        for (int k = 0; k < VPT; ++k) kh[k] += ex[k] * rz;  // last mask update is unused
      }
    }

    // ---- write khot row (coalesced b128 stores) ----
#pragma unroll
    for (int v = 0; v < NV4; ++v) {
      const int e = (v * BLOCK + t) * 4;
      float4 o;
      o.x = kh[4*v+0]; o.y = kh[4*v+1]; o.z = kh[4*v+2]; o.w = kh[4*v+3];
      *(float4*)(out + rbase + e) = o;
    }
  }
}

extern "C" void kernel_launch(void* const* d_in, const int* in_sizes, int n_in,
                              void* d_out, int out_size, void* d_ws, size_t ws_size,
                              hipStream_t stream) {
  (void)in_sizes; (void)n_in; (void)out_size; (void)d_ws; (void)ws_size;
  const float* scores = (const float*)d_in[0];
  const float* gnoise = (const float*)d_in[1];
  float* out = (float*)d_out;

  const size_t shmem = (size_t)2 * NCOLS * sizeof(float);  // 128 KB row stage
  subset_topk_kernel<<<dim3(NBLOCKS), dim3(BLOCK), shmem, stream>>>(scores, gnoise, out);
}